// Noise_70102456206131
// MI455X (gfx1250) — compile-verified
//
#include <hip/hip_runtime.h>

// out[r, c] = x[r, c] + ((u[c] <= 0.2) ? 0.5 * g[c] : 0)
//
// Pure streaming kernel: 822 MB total HBM traffic, ~35 us floor at 23.3 TB/s.
// float4 (b128) non-temporal loads/stores; noise held loop-invariant in
// registers via a compile-time grid stride that is a multiple of 196 float4
// column-groups, so each thread's column index never changes.

typedef float v4f __attribute__((ext_vector_type(4)));

#define COLS4   196u           // 784 / 4 float4 column-groups per row
#define BLOCK   256
#define GRID    1960           // 1960*256 = 501760 threads, 501760 % 196 == 0

static constexpr long STRIDE = (long)GRID * BLOCK;   // float4 units

__global__ __launch_bounds__(BLOCK) void Noise_70102456206131_kernel(
    const v4f* __restrict__ x,
    const float* __restrict__ u,
    const float* __restrict__ g,
    v4f* __restrict__ out,
    long total4)                      // total float4 elements = rows * 196
{
    const unsigned tid = blockIdx.x * BLOCK + threadIdx.x;  // < 501760, 32-bit
    long i = (long)tid;
    if (i >= total4) return;

    // Column group is invariant under i += STRIDE (STRIDE % 196 == 0):
    // compute the 4-wide noise vector once, keep it in VGPRs.
    const unsigned col4 = tid % COLS4;                // 32-bit mod, one-time
    const v4f u4 = *(const v4f*)(u + 4 * col4);       // 6 KB region -> cache
    const v4f g4 = *(const v4f*)(g + 4 * col4);
    v4f n;
    n.x = (u4.x <= 0.2f) ? 0.5f * g4.x : 0.0f;
    n.y = (u4.y <= 0.2f) ? 0.5f * g4.y : 0.0f;
    n.z = (u4.z <= 0.2f) ? 0.5f * g4.z : 0.0f;
    n.w = (u4.w <= 0.2f) ? 0.5f * g4.w : 0.0f;

    constexpr long S4 = 4 * STRIDE;

    // Main loop: 4 independent b128 NT loads in flight per thread, then
    // 4 b128 NT stores. Prefetch next stride-group into L2 (speculative,
    // silently dropped if past the end).
    for (; i + 3 * STRIDE < total4; i += S4) {
        __builtin_prefetch(&x[i + S4], 0, 0);
        v4f a = __builtin_nontemporal_load(&x[i]);
        v4f b = __builtin_nontemporal_load(&x[i + STRIDE]);
        v4f c = __builtin_nontemporal_load(&x[i + 2 * STRIDE]);
        v4f d = __builtin_nontemporal_load(&x[i + 3 * STRIDE]);
        a += n;
        b += n;
        c += n;
        d += n;
        __builtin_nontemporal_store(a, &out[i]);
        __builtin_nontemporal_store(b, &out[i + STRIDE]);
        __builtin_nontemporal_store(c, &out[i + 2 * STRIDE]);
        __builtin_nontemporal_store(d, &out[i + 3 * STRIDE]);
    }
    // Remainder
    for (; i < total4; i += STRIDE) {
        v4f a = __builtin_nontemporal_load(&x[i]);
        a += n;
        __builtin_nontemporal_store(a, &out[i]);
    }
}

extern "C" void kernel_launch(void* const* d_in, const int* in_sizes, int n_in,
                              void* d_out, int out_size, void* d_ws, size_t ws_size,
                              hipStream_t stream) {
    const float* x = (const float*)d_in[0];   // [rows, 784] f32
    const float* u = (const float*)d_in[1];   // [784] f32
    const float* g = (const float*)d_in[2];   // [784] f32
    float* out = (float*)d_out;

    const long total4 = (long)in_sizes[0] / 4;   // rows * 196

    Noise_70102456206131_kernel<<<GRID, BLOCK, 0, stream>>>(
        (const v4f*)x, u, g, (v4f*)out, total4);
}